// EqualtimeLayer_75488345194496
// MI455X (gfx1250) — compile-verified
//
#include <hip/hip_runtime.h>
#include <hip/hip_bf16.h>

// ---------------------------------------------------------------------------
// EqualtimeLayer forward for MI455X (gfx1250, wave32).
//
// B=32, PRE=1024, POST=1024. Grid: (POST/16, B) x 512 threads (16 wave32's).
// Per block (jt, b): stage spikes[b,:] + the [PRE x 16] delay/weight tile into
// LDS (transposed, bank-padded) -- via GLOBAL_LOAD_ASYNC_TO_LDS_B32 per-lane
// scatter when available (ASYNCcnt-tracked, overlapped with spike staging) --
// then wave w owns post neuron j = jt*16 + w: LDS bitonic sort of 1024
// (delayed-time, weight) pairs, prefix sums on the matrix pipe via chained
// V_WMMA_F32_16X16X4_F32 (scan == R x upper-triangular-ones), causal-window
// candidate solve, wave min-reduce.
// ---------------------------------------------------------------------------

typedef __attribute__((ext_vector_type(2))) float v2f;
typedef __attribute__((ext_vector_type(8))) float v8f;

namespace {
constexpr int kPre     = 1024;
constexpr int kPost    = 1024;
constexpr int kBatch   = 32;
constexpr int kJT      = 16;          // post neurons per block == waves per block
constexpr int kThreads = kJT * 32;    // 512
constexpr int kPad     = kPre + 1;    // 1025: bank-conflict padding (64 banks)
}

#if defined(__gfx1250__) &&                                              \
    __has_builtin(__builtin_amdgcn_global_load_async_to_lds_b32) &&      \
    __has_builtin(__builtin_amdgcn_s_wait_asynccnt)
#define EQT_ASYNC_LDS 1
#else
#define EQT_ASYNC_LDS 0
#endif

#if EQT_ASYNC_LDS
// Builtin prototype (from the round-2 diagnostic): param0 is AS1 int*
// (printed as '__device__ int *'), param1 AS3 int*, then imm offset, imm cpol.
// Generic->AS pointers via integer: AS1 keeps the 64-bit global address; for
// AS3 the low 32 bits of a generic LDS pointer are the LDS byte offset
// (flat-aperture rule).
#define EQT_GPTR(p) ((__attribute__((address_space(1))) int*)(uintptr_t)(p))
#define EQT_LPTR(p) ((__attribute__((address_space(3))) int*)(uintptr_t)(p))
#endif

// Compiler-reordering guard for same-wave LDS producer/consumer patterns.
// CDNA5 keeps DS ops from one wave in order (ISA ch.11), so no s_barrier is
// needed when only lanes of the same wave communicate.
__device__ __forceinline__ void wave_sync() {
  __builtin_amdgcn_fence(__ATOMIC_RELEASE, "workgroup");
  __builtin_amdgcn_wave_barrier();
  __builtin_amdgcn_fence(__ATOMIC_ACQUIRE, "workgroup");
}

extern "C" __global__ void __launch_bounds__(kThreads)
equaltime_fwd_kernel(const float* __restrict__ spikes,     // [B, PRE]
                     const float* __restrict__ weights,    // [PRE, POST]
                     const float* __restrict__ delays,     // [PRE, POST]
                     const float* __restrict__ thresholds, // [POST]
                     float* __restrict__ out)              // [B, POST]
{
  extern __shared__ float smem[];
  float* spk   = smem;                    // kPre
  float* keys  = spk + kPre;              // kJT * kPad   (delayed spike times)
  float* vals  = keys + kJT * kPad;       // kJT * kPad   (weights, sorted along)
  float* rs_w  = vals + kJT * kPad;       // kJT * 16     (WMMA row sums, w)
  float* rs_wt = rs_w + kJT * 16;         // kJT * 16     (WMMA row sums, w*t)

  const int tid  = threadIdx.x;
  const int lane = tid & 31;
  const int wave = tid >> 5;
  const int b    = blockIdx.y;
  const int j0   = blockIdx.x * kJT;

#if EQT_ASYNC_LDS
  // ---- async DMA the tile straight into LDS (transposed scatter) ----------
  // Each lane supplies its own LDS destination, so the transpose + bank
  // padding costs nothing. Tracked on ASYNCcnt, overlapped with spk staging.
  {
    const int c = tid & 15;
    for (int r = tid >> 4; r < kPre; r += (kThreads >> 4)) {
      const int g = r * kPost + j0 + c;
      __builtin_amdgcn_global_load_async_to_lds_b32(
          EQT_GPTR(delays + g),  EQT_LPTR(keys + c * kPad + r), 0, 0);
      __builtin_amdgcn_global_load_async_to_lds_b32(
          EQT_GPTR(weights + g), EQT_LPTR(vals + c * kPad + r), 0, 0);
    }
  }
  for (int i = tid; i < kPre; i += kThreads)        // overlapped with the DMA
    spk[i] = spikes[b * kPre + i];
  __builtin_amdgcn_s_wait_asynccnt(0);              // this wave's copies done
  __syncthreads();                                  // + everyone's spk stores
  // delays -> delayed spike times: each thread fixes the elements it DMA'd
  {
    const int c = tid & 15;
    for (int r = tid >> 4; r < kPre; r += (kThreads >> 4))
      keys[c * kPad + r] += spk[r];
  }
  __syncthreads();
#else
  // ---- synchronous staging fallback ---------------------------------------
  for (int i = tid; i < kPre; i += kThreads)
    spk[i] = spikes[b * kPre + i];
  __syncthreads();
  {
    const int c = tid & 15;
    for (int r = tid >> 4; r < kPre; r += (kThreads >> 4)) {
      const int g = r * kPost + j0 + c;
      if (r + 32 < kPre) {
        __builtin_prefetch(delays  + g + 32 * kPost, 0, 3);  // global_prefetch_b8
        __builtin_prefetch(weights + g + 32 * kPost, 0, 3);
      }
      keys[c * kPad + r] = spk[r] + delays[g];
      vals[c * kPad + r] = weights[g];
    }
  }
  __syncthreads();
#endif

  // ---- per-wave: bitonic sort of 1024 (key,val) pairs in LDS --------------
  float* kj = keys + wave * kPad;
  float* vj = vals + wave * kPad;

  for (unsigned k = 2; k <= (unsigned)kPre; k <<= 1) {
    for (unsigned s = k >> 1; s > 0; s >>= 1) {
      for (unsigned t = (unsigned)lane; t < (kPre >> 1); t += 32) {
        const unsigned i = ((t & ~(s - 1)) << 1) | (t & (s - 1));
        const unsigned p = i | s;
        const bool up = ((i & k) == 0);
        const float ki = kj[i], kp = kj[p];
        if (up ? (ki > kp) : (ki < kp)) {
          kj[i] = kp; kj[p] = ki;
          const float vi = vj[i], vp = vj[p];
          vj[i] = vp; vj[p] = vi;
        }
      }
      wave_sync();
    }
  }

  // ---- WMMA prefix sums + causal-window candidate search ------------------
  // Inclusive scan of a 16x16 value tile R as R x U (U = upper-tri ones),
  // built from 4 chained V_WMMA_F32_16X16X4_F32 (K=4 slices). Two scans
  // (w and w*t) share the B operand. Row/chunk carries stitched via LDS.
  const int jj = j0 + wave;
  const float theta = thresholds[jj];

  const int nCol  = lane & 15;      // C/D column, B column
  const int half  = lane >> 4;      // 0: rows 0-7 / K 0,1 ; 1: rows 8-15 / K 2,3
  const int khalf = half << 1;
  const int rbase = half << 3;
  float* rw  = rs_w  + wave * 16;
  float* rwt = rs_wt + wave * 16;

  float carry_w = 0.f, carry_wt = 0.f;
  float best = __builtin_inff();

  for (int chunk = 0; chunk < kPre / 256; ++chunk) {
    const int base = chunk * 256;
    v8f aw  = {};   // inclusive row-scan of w      (16x16 chunk)
    v8f awt = {};   // inclusive row-scan of w * t
#pragma unroll
    for (int kk = 0; kk < 4; ++kk) {
      const int c0 = 4 * kk + khalf;          // this lane's K columns: c0, c0+1
      const int m  = lane & 15;               // A-matrix row
      const int e0 = base + 16 * m + c0;
      float t0 = kj[e0], t1 = kj[e0 + 1];
      float w0 = vj[e0], w1 = vj[e0 + 1];
      if (t0 == __builtin_inff()) { w0 = 0.f; t0 = 0.f; }   // inf-event mask
      if (t1 == __builtin_inff()) { w1 = 0.f; t1 = 0.f; }
      v2f a_w;  a_w.x  = w0;      a_w.y  = w1;
      v2f a_wt; a_wt.x = w0 * t0; a_wt.y = w1 * t1;
      v2f bu;   bu.x = (c0     <= nCol) ? 1.f : 0.f;        // rows of U
                bu.y = (c0 + 1 <= nCol) ? 1.f : 0.f;
      aw  = __builtin_amdgcn_wmma_f32_16x16x4_f32(false, a_w,  false, bu,
                                                  (short)0, aw,  false, false);
      awt = __builtin_amdgcn_wmma_f32_16x16x4_f32(false, a_wt, false, bu,
                                                  (short)0, awt, false, false);
    }

    // publish per-row totals (column 15) so every lane can build offsets
    if (nCol == 15) {
#pragma unroll
      for (int q = 0; q < 8; ++q) { rw[rbase + q] = aw[q]; rwt[rbase + q] = awt[q]; }
    }
    wave_sync();

    float totw = 0.f, totwt = 0.f;
#pragma unroll
    for (int r = 0; r < 16; ++r) { totw += rw[r]; totwt += rwt[r]; }

    float sw = carry_w, swt = carry_wt;       // exclusive offset for row rbase
    if (half) {
#pragma unroll
      for (int r = 0; r < 8; ++r) { sw += rw[r]; swt += rwt[r]; }
    }

#pragma unroll
    for (int q = 0; q < 8; ++q) {
      const int r = rbase + q;
      const int i = base + 16 * r + nCol;     // sorted event index
      const float cw  = aw[q]  + sw;          // cumw[i]
      const float cwt = awt[q] + swt;         // cumwt[i]
      const float cand = (cw > 0.f) ? ((theta + cwt) / cw) : __builtin_inff();
      const float ti = kj[i];
      bool bad = (cand < ti);                       // fires before window opens
      if (i < kPre - 1) bad = bad || (cand > kj[i + 1]);  // fires after next event
      if (!bad) best = fminf(best, cand);
      sw  += rw[r];                            // advance to next row's offset
      swt += rwt[r];
    }
    carry_w  += totw;
    carry_wt += totwt;
    wave_sync();                               // before next chunk rewrites rw/rwt
  }

  // ---- wave32 min-reduce, lane 0 writes the output ------------------------
#pragma unroll
  for (int off = 16; off > 0; off >>= 1)
    best = fminf(best, __shfl_xor(best, off, 32));
  if (lane == 0) out[b * kPost + jj] = best;
}

extern "C" void kernel_launch(void* const* d_in, const int* in_sizes, int n_in,
                              void* d_out, int out_size, void* d_ws, size_t ws_size,
                              hipStream_t stream) {
  (void)in_sizes; (void)n_in; (void)out_size; (void)d_ws; (void)ws_size;
  const float* spikes     = (const float*)d_in[0];
  const float* weights    = (const float*)d_in[1];
  const float* delays     = (const float*)d_in[2];
  const float* thresholds = (const float*)d_in[3];
  float* out = (float*)d_out;

  dim3 grid(kPost / kJT, kBatch);   // (64, 32)
  dim3 block(kThreads);             // 512 = 16 wave32
  const size_t shmem =
      (size_t)(kPre + 2 * kJT * kPad + 2 * kJT * 16) * sizeof(float); // ~134 KB

  equaltime_fwd_kernel<<<grid, block, shmem, stream>>>(
      spikes, weights, delays, thresholds, out);
}